// MessageUpdate_44367012168460
// MI455X (gfx1250) — compile-verified
//
#include <hip/hip_runtime.h>
#include <hip/hip_bf16.h>

// MI455X / gfx1250, wave32. f32 WMMA (V_WMMA_F32_16X16X4_F32) +
// double-buffered GLOBAL_LOAD_ASYNC_TO_LDS_B128 staging.

typedef __attribute__((ext_vector_type(2))) float v2f;
typedef __attribute__((ext_vector_type(8))) float v8f;

#define N_SITES 64
#define N_EDGES 8192
#define BATCH   64
#define IN_F    64
#define BOND_F  32
#define F_CAT   160
#define OUT_F   64
#define KSTEPS  40          // 160 / 4
#define VSTRIDE 164         // padded LDS row stride (floats); 656 B, 16B-aligned
#define ESPLIT  4
#define ERANGE  (N_EDGES / ESPLIT)   // 2048

// flat pointer -> 32-bit LDS offset (flat LDS aperture keeps the wave-relative
// offset in addr[31:0], per CDNA5 ISA 10.2 aperture rules)
__device__ __forceinline__ unsigned lds_off(const void* p) {
    return (unsigned)(unsigned long long)(uintptr_t)p;
}

// ---------------------------------------------------------------------------
// Kernel 1: gather per-site weight matrices into a WMMA-friendly layout:
// Wpack[s][k/2][n][2]  (K-pairs adjacent -> b64 B-fragment loads), plus bias.
// ---------------------------------------------------------------------------
__global__ void pack_weights(const float* __restrict__ W,      // (OUT_F, F_CAT, n_pair)
                             const float* __restrict__ bparam, // (OUT_F, n_one)
                             const int*   __restrict__ diag_oid,
                             const int*   __restrict__ bias_oid,
                             int n_pair, int n_one,
                             float* __restrict__ Wpack,  // (64, 80, 64, 2)
                             float* __restrict__ bpack)  // (64, 64)
{
    const int s = blockIdx.x;
    const int g = diag_oid[s];
    const int h = bias_oid[s];
    for (int idx = threadIdx.x; idx < F_CAT * OUT_F; idx += blockDim.x) {
        const int kp  = idx >> 7;        // K-pair 0..79
        const int rem = idx & 127;
        const int n   = rem >> 1;        // 0..63
        const int i   = rem & 1;         // low/high of pair
        Wpack[s * (F_CAT * OUT_F) + idx] = W[(n * F_CAT + 2 * kp + i) * n_pair + g];
    }
    if (threadIdx.x < OUT_F)
        bpack[s * OUT_F + threadIdx.x] = bparam[threadIdx.x * n_one + h];
}

// ---------------------------------------------------------------------------
// Kernel 2: site-major edge GEMM with double-buffered async-to-LDS staging.
// Block = (site s, 32-batch chunk, edge-quarter j).
// ---------------------------------------------------------------------------
__launch_bounds__(256)
__global__ void edge_gemm(const float* __restrict__ sites,  // (B, 64, 64)
                          const float* __restrict__ bonds,  // (B, 8192, 32)
                          const float* __restrict__ Wpack,  // (64, 80, 64, 2)
                          const float* __restrict__ bpack,  // (64, 64)
                          const float* __restrict__ Watt,   // (64)
                          const float* __restrict__ batt,   // (1)
                          const int*   __restrict__ idx1,
                          const int*   __restrict__ idx2v,
                          float* __restrict__ part)         // (ESPLIT, B, 64, 64)
{
    __shared__ float VsA[32 * VSTRIDE];  // double-buffered V tiles
    __shared__ float VsB[32 * VSTRIDE];
    __shared__ float Lat[32 * 65];
    __shared__ float AttL[32];
    __shared__ float BiasL[OUT_F];
    __shared__ float WattL[OUT_F];
    __shared__ float battL;
    __shared__ int   elist[ERANGE];
    __shared__ int   wcnt[8];
    __shared__ int   ecnt;

    const int s   = blockIdx.x;
    const int b0  = blockIdx.y * 32;
    const int j   = blockIdx.z;
    const int tid = threadIdx.x;
    const int lane = tid & 31;
    const int wid  = tid >> 5;

    if (tid < OUT_F) { BiasL[tid] = bpack[s * OUT_F + tid]; WattL[tid] = Watt[tid]; }
    if (tid == 0)    { battL = batt[0]; ecnt = 0; }
    __syncthreads();

    // ---- deterministic compaction of edges with idx2[e] == s in this quarter
    for (int c = 0; c < ERANGE / 256; ++c) {
        const int e = j * ERANGE + c * 256 + tid;
        const bool m = (idx2v[e] == s);
        const unsigned long long bal = __ballot(m);
        if (lane == 0) wcnt[wid] = __popcll(bal);
        __syncthreads();
        int off = ecnt;
        for (int w = 0; w < wid; ++w) off += wcnt[w];
        if (m) elist[off + __popcll(bal & ((1ull << lane) - 1ull))] = e;
        __syncthreads();
        if (tid == 0) { int t2 = 0; for (int w = 0; w < 8; ++w) t2 += wcnt[w]; ecnt += t2; }
        __syncthreads();
    }
    const int nE = ecnt;

    // ---- async staging of one V tile: 5 x B128 per thread, no VGPR round-trip
    const unsigned vb0 = lds_off(VsA);
    const unsigned vb1 = lds_off(VsB);
    const int sb = tid >> 3, sq = tid & 7;        // staging row / chunk lane
    const int sgb = b0 + sb;
    const float* site_row = sites + sgb * (N_SITES * IN_F);
    const float* bond_row = bonds + (size_t)sgb * N_EDGES * BOND_F;
    const unsigned q16 = (unsigned)sq * 16u;

    auto stage = [&](int e, int s1, unsigned vbase) {
        const unsigned l0 = vbase + (unsigned)(sb * VSTRIDE * 4);
        const unsigned long long g1 = (unsigned long long)(uintptr_t)(site_row + s1 * IN_F);
        const unsigned long long g2 = (unsigned long long)(uintptr_t)(site_row + s  * IN_F);
        const unsigned long long gb = (unsigned long long)(uintptr_t)(bond_row + (size_t)e * BOND_F);
        asm volatile("global_load_async_to_lds_b128 %0, %1, off"
                     :: "v"(l0 + q16),        "v"(g1 + q16)        : "memory");
        asm volatile("global_load_async_to_lds_b128 %0, %1, off"
                     :: "v"(l0 + 128 + q16),  "v"(g1 + 128 + q16)  : "memory");
        asm volatile("global_load_async_to_lds_b128 %0, %1, off"
                     :: "v"(l0 + 256 + q16),  "v"(g2 + q16)        : "memory");
        asm volatile("global_load_async_to_lds_b128 %0, %1, off"
                     :: "v"(l0 + 384 + q16),  "v"(g2 + 128 + q16)  : "memory");
        asm volatile("global_load_async_to_lds_b128 %0, %1, off"
                     :: "v"(l0 + 512 + q16),  "v"(gb + q16)        : "memory");
    };

    // ---- per-wave WMMA tile assignment: 2 M-tiles x 4 N-tiles (8 waves)
    const int mt = wid >> 2, nt = wid & 3;
    const int m0 = mt * 16, n0 = nt * 16;
    const int ll = lane & 15, lh = lane >> 4;

    // prologue: kick off first tile's async stage, then load B fragments
    // (site-invariant) while it is in flight.
    if (nE > 0) stage(elist[0], idx1[elist[0]], vb0);

    // B layout (4x16 KxN): V0 = rows K0 (lanes 0-15)/K2 (lanes 16-31), V1 = K1/K3.
    // Wpack pairs (k,k+1) at fixed n -> one b64 per K-step.
    const float* __restrict__ Wsg = Wpack + s * (F_CAT * OUT_F) + lh * 128 + (n0 + ll) * 2;
    v2f Bf[KSTEPS];
#pragma unroll
    for (int t = 0; t < KSTEPS; ++t) Bf[t] = *(const v2f*)(Wsg + t * 256);

    v8f acc = {};
    const float* Ab0 = &VsA[(m0 + ll) * VSTRIDE + lh * 2];
    const float* Ab1 = &VsB[(m0 + ll) * VSTRIDE + lh * 2];

    for (int ei = 0; ei < nE; ++ei) {
        const int cur = ei & 1;
        if (ei + 1 < nE) {
            const int en = elist[ei + 1];
            stage(en, idx1[en], cur ? vb0 : vb1);   // fill the other buffer
            // async loads complete in order: <=5 outstanding => current tile landed
            asm volatile("s_wait_asynccnt 0x5" ::: "memory");
        } else {
            asm volatile("s_wait_asynccnt 0x0" ::: "memory");
        }
        __syncthreads();

        // K-loop: A from LDS (ds_load_b64 pairs), B from registers, 40 WMMAs.
        const float* Ab = cur ? Ab1 : Ab0;
        v8f c = {};
#pragma unroll
        for (int t = 0; t < KSTEPS; ++t) {
            v2f a = *(const v2f*)(Ab + t * 4);
            c = __builtin_amdgcn_wmma_f32_16x16x4_f32(false, a, false, Bf[t],
                                                      (short)0, c, false, false);
        }

        // bias + leaky relu; D layout: reg r -> M = m0 + r + 8*lh, N = n0 + ll
        const float bn = BiasL[n0 + ll];
        float lat8[8];
#pragma unroll
        for (int r = 0; r < 8; ++r) {
            float x = c[r] + bn;
            x = (x >= 0.0f) ? x : 0.01f * x;
            lat8[r] = x;
            Lat[(m0 + r + 8 * lh) * 65 + n0 + ll] = x;
        }
        __syncthreads();

        // attention: per batch row b, sigmoid(sum_n Lat[b][n]*Watt[n] + batt)
        {
            const int b = tid >> 3, q = tid & 7;
            const float* lr = &Lat[b * 65 + q * 8];
            float p = 0.0f;
#pragma unroll
            for (int i2 = 0; i2 < 8; ++i2) p += lr[i2] * WattL[q * 8 + i2];
            p += __shfl_xor(p, 1);
            p += __shfl_xor(p, 2);
            p += __shfl_xor(p, 4);
            if (q == 0) AttL[b] = 1.0f / (1.0f + __expf(-(p + battL)));
        }
        __syncthreads();

        // gate and accumulate over edges (registers only)
#pragma unroll
        for (int r = 0; r < 8; ++r)
            acc[r] += AttL[m0 + r + 8 * lh] * lat8[r];
        // no trailing barrier: the two barriers above fence both the V-buffer
        // refill (issued 2 barriers after the last read of that buffer) and Lat/AttL.
    }

    // write this split's partial: part[j][gb][s][n] (every element written,
    // including zeros when nE==0 -> no buffer zero-init needed)
#pragma unroll
    for (int r = 0; r < 8; ++r) {
        const int gb = b0 + m0 + r + 8 * lh;
        part[((j * BATCH + gb) * N_SITES + s) * OUT_F + n0 + ll] = acc[r];
    }
}

// ---------------------------------------------------------------------------
// Kernel 3: sum the 4 edge-split partials into d_out (deterministic order).
// ---------------------------------------------------------------------------
__global__ void reduce_parts(const float* __restrict__ part, float* __restrict__ out)
{
    const int i = blockIdx.x * 256 + threadIdx.x;
    const int S = BATCH * N_SITES * OUT_F;   // 262144
    out[i] = part[i] + part[i + S] + part[i + 2 * S] + part[i + 3 * S];
}

// ---------------------------------------------------------------------------
extern "C" void kernel_launch(void* const* d_in, const int* in_sizes, int n_in,
                              void* d_out, int out_size, void* d_ws, size_t ws_size,
                              hipStream_t stream)
{
    const float* sites  = (const float*)d_in[0];
    const float* bonds  = (const float*)d_in[1];
    const float* Wparam = (const float*)d_in[2];
    const float* bparam = (const float*)d_in[3];
    const float* Watt   = (const float*)d_in[4];
    const float* batt   = (const float*)d_in[5];
    const int*   idx1   = (const int*)d_in[6];
    const int*   idx2   = (const int*)d_in[7];
    const int*   diag   = (const int*)d_in[8];
    const int*   biasO  = (const int*)d_in[9];

    const int n_pair = in_sizes[2] / (OUT_F * F_CAT);
    const int n_one  = in_sizes[3] / OUT_F;

    float* ws    = (float*)d_ws;
    float* Wpack = ws;                                         // 64*160*64 floats
    float* bpack = ws + N_SITES * F_CAT * OUT_F;               // 64*64 floats
    float* part  = bpack + N_SITES * OUT_F;                    // 4*262144 floats

    pack_weights<<<N_SITES, 256, 0, stream>>>(Wparam, bparam, diag, biasO,
                                              n_pair, n_one, Wpack, bpack);

    dim3 grid(N_SITES, BATCH / 32, ESPLIT);   // 64 x 2 x 4 = 512 blocks, 8 waves each
    edge_gemm<<<grid, 256, 0, stream>>>(sites, bonds, Wpack, bpack, Watt, batt,
                                        idx1, idx2, part);

    reduce_parts<<<(BATCH * N_SITES * OUT_F) / 256, 256, 0, stream>>>(part, (float*)d_out);
}